// MambaBackbone_49770081026634
// MI455X (gfx1250) — compile-verified
//
#include <hip/hip_runtime.h>
#include <hip/hip_bf16.h>

// ---------------------------------------------------------------------------
// Mamba backbone for MI455X (gfx1250, wave32).
//   B=4, L=2048, D_MODEL=128, D_INNER=256, D_STATE=16, D_CONV=4, DT_RANK=8, NL=6
// GEMMs run on V_WMMA_F32_16X16X4_F32 (fp32 matrix pipe, matches reference
// precision). Scan is 4096-lane parallel (lane = (b, c, state-quad)) with
// software-pipelined loads and shfl reductions.
// ---------------------------------------------------------------------------

typedef __attribute__((ext_vector_type(2))) float v2f;
typedef __attribute__((ext_vector_type(4))) float v4f;
typedef __attribute__((ext_vector_type(8))) float v8f;

#define BB     4
#define LL     2048
#define DM     128
#define DI     256
#define DS     16
#define DTRK   8
#define ROWS   (BB * LL)          // 8192
#define NPROJ  (DTRK + 2 * DS)    // 40

__device__ __forceinline__ float silu_f(float x) {
    return x * (1.0f / (1.0f + __expf(-x)));
}

// ---------------------------------------------------------------------------
// WMMA GEMM:  C[M x Nact] = A[M x K] * W^T   (W is [N x K] row-major)
// Wave tile = (MT*16) x (NT*16). fp32 16x16x4 WMMA, K-loop step 4.
// Fragment layouts per CDNA5 ISA 7.12.2:
//   A: lane m = lane%16, a[v] = A[m][2*(lane/16)+v]
//   B: lane n = lane%16, b[v] = B[2*(lane/16)+v][n] = W[n][2*(lane/16)+v]
//   D: lane n = lane%16, d[v] = D[v + 8*(lane/16)][n]
// ---------------------------------------------------------------------------
template <int MT, int NT, bool NBOUND>
__global__ __launch_bounds__(256) void gemm_wmma_kernel(
    const float* __restrict__ A, const float* __restrict__ W,
    float* __restrict__ C, int M, int K, int ldc, int Nact, int nTilesN)
{
    const int wave   = threadIdx.x >> 5;
    const int lane   = threadIdx.x & 31;
    const int lane16 = lane & 15;
    const int half   = lane >> 4;

    const int wg = blockIdx.x * 8 + wave;
    const int tN = wg % nTilesN;
    const int tM = wg / nTilesN;
    const int m0 = tM * (MT * 16);
    const int n0 = tN * (NT * 16);
    if (m0 >= M) return;

    v8f acc[MT][NT];
#pragma unroll
    for (int i = 0; i < MT; ++i)
#pragma unroll
        for (int j = 0; j < NT; ++j)
            acc[i][j] = v8f{0.f, 0.f, 0.f, 0.f, 0.f, 0.f, 0.f, 0.f};

    const int kOff = half * 2;

    const float* aRow[MT];
#pragma unroll
    for (int i = 0; i < MT; ++i)
        aRow[i] = A + (size_t)(m0 + i * 16 + lane16) * K + kOff;

    const float* bRow[NT];
    float        bMask[NT];
#pragma unroll
    for (int j = 0; j < NT; ++j) {
        int n     = n0 + j * 16 + lane16;
        bool ok   = (!NBOUND) || (n < Nact);
        bMask[j]  = ok ? 1.0f : 0.0f;
        bRow[j]   = W + (size_t)(ok ? n : 0) * K + kOff;
    }

#pragma unroll 4
    for (int kk = 0; kk < K; kk += 4) {
        v2f aF[MT], bF[NT];
#pragma unroll
        for (int i = 0; i < MT; ++i)
            aF[i] = *(const v2f*)(aRow[i] + kk);
#pragma unroll
        for (int j = 0; j < NT; ++j) {
            v2f t = *(const v2f*)(bRow[j] + kk);
            if (NBOUND) t *= bMask[j];
            bF[j] = t;
        }
#pragma unroll
        for (int i = 0; i < MT; ++i)
#pragma unroll
            for (int j = 0; j < NT; ++j)
                acc[i][j] = __builtin_amdgcn_wmma_f32_16x16x4_f32(
                    false, aF[i], false, bF[j], (short)0, acc[i][j],
                    false, false);
    }

#pragma unroll
    for (int i = 0; i < MT; ++i) {
#pragma unroll
        for (int j = 0; j < NT; ++j) {
            int n = n0 + j * 16 + lane16;
            if (NBOUND && n >= Nact) continue;
#pragma unroll
            for (int v = 0; v < 8; ++v) {
                int m = m0 + i * 16 + half * 8 + v;
                C[(size_t)m * ldc + n] = acc[i][j][v];
            }
        }
    }
}

// ---------------------------------------------------------------------------
// Causal depthwise conv (k=4) + bias + SiLU.  xz is (ROWS, 512); xb = cols
// [0,256).  Output xconv (ROWS, 256).
// ---------------------------------------------------------------------------
__global__ __launch_bounds__(256) void conv_silu_kernel(
    const float* __restrict__ xz, const float* __restrict__ cw,
    const float* __restrict__ cb, float* __restrict__ xconv)
{
    int idx = blockIdx.x * blockDim.x + threadIdx.x;   // over ROWS*DI
    int c   = idx & (DI - 1);
    int bl  = idx >> 8;                                // b*L + t
    int t   = bl & (LL - 1);

    float acc = cb[c];
#pragma unroll
    for (int j = 0; j < 4; ++j) {
        int tt = t - 3 + j;
        if (tt >= 0)
            acc += xz[(size_t)(bl - 3 + j) * (2 * DI) + c] * cw[c * 4 + j];
    }
    xconv[idx] = silu_f(acc);
}

// ---------------------------------------------------------------------------
// dt = softplus(proj[:, :8] @ Wdt^T + bdt)   -> (ROWS, 256)
// ---------------------------------------------------------------------------
__global__ __launch_bounds__(256) void dtproj_kernel(
    const float* __restrict__ proj, const float* __restrict__ Wdt,
    const float* __restrict__ bdt, float* __restrict__ dt)
{
    int idx = blockIdx.x * blockDim.x + threadIdx.x;   // over ROWS*DI
    int c   = idx & (DI - 1);
    int m   = idx >> 8;

    const float* p = proj + (size_t)m * NPROJ;
    const float* w = Wdt + c * DTRK;
    float acc = bdt[c];
#pragma unroll
    for (int r = 0; r < DTRK; ++r) acc += p[r] * w[r];
    dt[idx] = (acc > 20.f) ? acc : log1pf(__expf(acc));
}

// ---------------------------------------------------------------------------
// Selective scan + D skip + SiLU(z) gating.
// Lane = (b, c, state-quad): 4096 lanes.  4 states per lane in registers,
// next-timestep loads issued before current compute (software pipeline),
// y reduced over the 4 state-lanes with shfl_xor.
// ---------------------------------------------------------------------------
__global__ __launch_bounds__(256) void scan_kernel(
    const float* __restrict__ dt, const float* __restrict__ xconv,
    const float* __restrict__ xz, const float* __restrict__ proj,
    const float* __restrict__ A_log, const float* __restrict__ Dp,
    float* __restrict__ Y)
{
    int tid = blockIdx.x * blockDim.x + threadIdx.x;   // 0..4095
    int sg  = tid & 3;                                  // state quad
    int c   = (tid >> 2) & (DI - 1);
    int b   = tid >> 10;
    int sg4 = sg * 4;

    float nA[4], h[4];
#pragma unroll
    for (int j = 0; j < 4; ++j) {
        nA[j] = -__expf(A_log[c * DS + sg4 + j]);
        h[j]  = 0.0f;
    }
    const float Dv   = Dp[c];
    const size_t base = (size_t)b * LL;

    // prologue loads (t = 0)
    size_t r0 = base;
    float dC = dt[r0 * DI + c];
    float xC = xconv[r0 * DI + c];
    float zC = xz[r0 * (2 * DI) + DI + c];
    v4f   Bc = *(const v4f*)(proj + r0 * NPROJ + DTRK + sg4);
    v4f   Cc = *(const v4f*)(proj + r0 * NPROJ + DTRK + DS + sg4);

    for (int t = 0; t < LL; ++t) {
        // issue next-step loads early (independent of compute below)
        int    tn = (t + 1 < LL) ? (t + 1) : t;
        size_t rn = base + tn;
        float dN = dt[rn * DI + c];
        float xN = xconv[rn * DI + c];
        float zN = xz[rn * (2 * DI) + DI + c];
        v4f   Bn = *(const v4f*)(proj + rn * NPROJ + DTRK + sg4);
        v4f   Cn = *(const v4f*)(proj + rn * NPROJ + DTRK + DS + sg4);
        if (sg == 0 && t + 8 < LL)
            __builtin_prefetch(proj + (base + t + 8) * NPROJ, 0, 1);

        float bx = dC * xC;
        float yp = 0.0f;
#pragma unroll
        for (int j = 0; j < 4; ++j) {
            float a = __expf(dC * nA[j]);
            h[j] = a * h[j] + bx * Bc[j];
            yp  += h[j] * Cc[j];
        }
        yp += __shfl_xor(yp, 1, 32);
        yp += __shfl_xor(yp, 2, 32);
        if (sg == 0)
            Y[(base + t) * DI + c] = (yp + xC * Dv) * silu_f(zC);

        dC = dN; xC = xN; zC = zN; Bc = Bn; Cc = Cn;
    }
}

// ---------------------------------------------------------------------------
// out = LayerNorm(xin [+ resid]) * g + b.  One wave per 128-wide row,
// wave32 butterfly reductions.
// ---------------------------------------------------------------------------
__global__ __launch_bounds__(256) void resid_ln_kernel(
    const float* __restrict__ xin, const float* __restrict__ resid,
    const float* __restrict__ g, const float* __restrict__ bb,
    float* __restrict__ out, int rows)
{
    int row  = (blockIdx.x * blockDim.x + threadIdx.x) >> 5;
    int lane = threadIdx.x & 31;
    if (row >= rows) return;

    const float* xr = xin + (size_t)row * DM;
    float v[4], sum = 0.f, sq = 0.f;
#pragma unroll
    for (int j = 0; j < 4; ++j) {
        float t = xr[lane + j * 32];
        if (resid) t += resid[(size_t)row * DM + lane + j * 32];
        v[j] = t; sum += t; sq += t * t;
    }
#pragma unroll
    for (int off = 16; off >= 1; off >>= 1) {
        sum += __shfl_xor(sum, off, 32);
        sq  += __shfl_xor(sq,  off, 32);
    }
    float mu  = sum * (1.0f / DM);
    float var = sq * (1.0f / DM) - mu * mu;
    float rs  = rsqrtf(var + 1e-5f);
#pragma unroll
    for (int j = 0; j < 4; ++j) {
        int d = lane + j * 32;
        out[(size_t)row * DM + d] = (v[j] - mu) * rs * g[d] + bb[d];
    }
}

// ---------------------------------------------------------------------------
extern "C" void kernel_launch(void* const* d_in, const int* in_sizes, int n_in,
                              void* d_out, int out_size, void* d_ws,
                              size_t ws_size, hipStream_t stream)
{
    const float* x        = (const float*)d_in[0];
    const float* in_proj  = (const float*)d_in[1];   // (NL, 512, 128)
    const float* conv_w   = (const float*)d_in[2];   // (NL, 256, 4)
    const float* conv_b   = (const float*)d_in[3];   // (NL, 256)
    const float* xproj_w  = (const float*)d_in[4];   // (NL, 40, 256)
    const float* dt_w     = (const float*)d_in[5];   // (NL, 256, 8)
    const float* dt_b     = (const float*)d_in[6];   // (NL, 256)
    const float* A_log    = (const float*)d_in[7];   // (NL, 256, 16)
    const float* Dp       = (const float*)d_in[8];   // (NL, 256)
    const float* out_w    = (const float*)d_in[9];   // (NL, 128, 256)
    const float* ln_g     = (const float*)d_in[10];  // (NL, 128)
    const float* ln_b     = (const float*)d_in[11];  // (NL, 128)
    const float* outln_g  = (const float*)d_in[12];  // (128)
    const float* outln_b  = (const float*)d_in[13];  // (128)

    float* ws = (float*)d_ws;
    size_t off = 0;
    float* h     = ws + off; off += (size_t)ROWS * DM;        // residual stream
    float* xzb   = ws + off; off += (size_t)ROWS * 2 * DI;    // in_proj out
    float* xcv   = ws + off; off += (size_t)ROWS * DI;        // conv+silu
    float* prj   = ws + off; off += (size_t)ROWS * NPROJ;     // x_proj out
    float* dtb   = ws + off; off += (size_t)ROWS * DI;        // dt
    float* yb    = ws + off; off += (size_t)ROWS * DI;        // gated scan out
    float* tmp   = ws + off; off += (size_t)ROWS * DM;        // out_proj out

    hipMemcpyAsync(h, x, (size_t)ROWS * DM * sizeof(float),
                   hipMemcpyDeviceToDevice, stream);

    for (int l = 0; l < 6; ++l) {
        const float* Wi  = in_proj + (size_t)l * 2 * DI * DM;
        const float* cw  = conv_w  + (size_t)l * DI * 4;
        const float* cb  = conv_b  + (size_t)l * DI;
        const float* Wx  = xproj_w + (size_t)l * NPROJ * DI;
        const float* Wdt = dt_w    + (size_t)l * DI * DTRK;
        const float* bdt = dt_b    + (size_t)l * DI;
        const float* Al  = A_log   + (size_t)l * DI * DS;
        const float* Dl  = Dp      + (size_t)l * DI;
        const float* Wo  = out_w   + (size_t)l * DM * DI;
        const float* lg  = ln_g    + (size_t)l * DM;
        const float* lb  = ln_b    + (size_t)l * DM;

        // in_proj: M=8192, N=512, K=128 -> 32x32 wave tiles, 16 N-tiles
        gemm_wmma_kernel<2, 2, false><<<512, 256, 0, stream>>>(
            h, Wi, xzb, ROWS, DM, 2 * DI, 2 * DI, 16);

        conv_silu_kernel<<<(ROWS * DI) / 256, 256, 0, stream>>>(
            xzb, cw, cb, xcv);

        // x_proj: M=8192, Nact=40 (pad 48), K=256 -> 32x48 tiles, 1 N-tile
        gemm_wmma_kernel<2, 3, true><<<32, 256, 0, stream>>>(
            xcv, Wx, prj, ROWS, DI, NPROJ, NPROJ, 1);

        dtproj_kernel<<<(ROWS * DI) / 256, 256, 0, stream>>>(
            prj, Wdt, bdt, dtb);

        scan_kernel<<<16, 256, 0, stream>>>(
            dtb, xcv, xzb, prj, Al, Dl, yb);

        // out_proj: M=8192, N=128, K=256 -> 32x32 tiles, 4 N-tiles
        gemm_wmma_kernel<2, 2, false><<<128, 256, 0, stream>>>(
            yb, Wo, tmp, ROWS, DI, DM, DM, 4);

        // h = LN(tmp + h)
        resid_ln_kernel<<<ROWS / 8, 256, 0, stream>>>(
            tmp, h, lg, lb, h, ROWS);
    }

    // final LN -> d_out
    resid_ln_kernel<<<ROWS / 8, 256, 0, stream>>>(
        h, nullptr, outln_g, outln_b, (float*)d_out, ROWS);
}